// RNN_18597208391864
// MI455X (gfx1250) — compile-verified
//
// CDNA5 / gfx1250 RNN: out[b,t,:] = h_t,  h_t = (x_t @ Wx^T + b) + h_{t-1} @ Wh^T
// B=16, T=512, H=2048, f32.
//
// Phase 1: u = x @ Wx^T + b   -> V_WMMA_F32_16X16X4_F32, TDM-staged LDS tiles,
//                                double buffered with s_wait_tensorcnt.
// Phase 2: sequential scan    -> Wh tile resident in LDS (loaded once via TDM),
//                                per-step TDM stage of h_{t-1}, per-step cross-WG
//                                release/acquire arrive-counter sync.
#include <hip/hip_runtime.h>
#include <stdint.h>

typedef __attribute__((ext_vector_type(2))) float v2f;
typedef __attribute__((ext_vector_type(8))) float v8f;
typedef __attribute__((ext_vector_type(4))) unsigned int v4u;
typedef __attribute__((ext_vector_type(4))) int v4i;
typedef __attribute__((ext_vector_type(8))) int v8i;

#define B_DIM 16
#define T_DIM 512
#define H_DIM 2048
#define WROW2 4096   // row stride of W (H x 2H, row-major); Wx=W[g][h], Wh=W[g][2048+h]

// ---- TDM availability ------------------------------------------------------
#if defined(__has_builtin)
# if __has_builtin(__builtin_amdgcn_tensor_load_to_lds)
#  define USE_TDM 1
# endif
#endif
#ifndef USE_TDM
# define USE_TDM 0
#endif

#if USE_TDM
// Issue one 2D TENSOR_LOAD_TO_LDS. D# packing per CDNA5 ISA §8 (2D case,
// groups 2/3 zero => <=2D). lds_byte_off assumed relative to dynamic-LDS base 0.
// pad_en inserts pad_amount(+1) dwords every 2^(pad_ic+1) dwords (load only).
__device__ __forceinline__ void tdm_load_2d(uint32_t lds_byte_off, const float* gsrc,
                                            uint32_t tile_w, uint32_t tile_h,
                                            uint64_t row_stride_elems,
                                            uint32_t pad_ic, uint32_t pad_ac) {
  uint64_t ga = (uint64_t)(uintptr_t)gsrc;
  v4u g0;
  g0.x = 1u;                                             // count=1, user mode
  g0.y = lds_byte_off;                                   // lds_addr
  g0.z = (uint32_t)ga;                                   // global_addr[31:0]
  g0.w = (uint32_t)((ga >> 32) & 0x1FFFFFFu) | (2u << 30); // addr[56:32] | type=2
  const uint32_t tdim0 = 0x40000000u;                    // huge => no OOB clip
  const uint32_t tdim1 = 0x00100000u;
  uint64_t s0 = row_stride_elems;                        // tensor_dim0_stride
  v8i g1;
  g1[0] = (int)((2u << 16) |                             // data_size = 4B
                (1u << 20) | (pad_ic << 22) | (pad_ac << 25)); // pad enable
  g1[1] = (int)((tdim0 & 0xFFFFu) << 16);                // tensor_dim0 lo16
  g1[2] = (int)((tdim0 >> 16) | ((tdim1 & 0xFFFFu) << 16));
  g1[3] = (int)((tdim1 >> 16) | (tile_w << 16));         // tile_dim0
  g1[4] = (int)(tile_h & 0xFFFFu);                       // tile_dim1 (tile_dim2=0)
  g1[5] = (int)(uint32_t)(s0 & 0xFFFFFFFFu);
  g1[6] = (int)(uint32_t)((s0 >> 32) & 0xFFFFu);         // dim1_stride unused (2D)
  g1[7] = 0;
  v4i gz = {0, 0, 0, 0};
#if __clang_major__ >= 23
  v8i gz8 = {0, 0, 0, 0, 0, 0, 0, 0};
  __builtin_amdgcn_tensor_load_to_lds(g0, g1, gz, gz, gz8, 0);
#else
  __builtin_amdgcn_tensor_load_to_lds(g0, g1, gz, gz, 0);
#endif
}
#endif // USE_TDM

// ---- WMMA wrapper ----------------------------------------------------------
// D(16x16 f32) = A(16x4 f32) * B(4x16 f32) + C. A/B: 2 VGPRs/lane.
__device__ __forceinline__ v8f wmma_f32(v2f a, v2f b, v8f c) {
  return __builtin_amdgcn_wmma_f32_16x16x4_f32(false, a, false, b, (short)0, c,
                                               false, false);
}

// ============================================================================
// Kernel A: u = x @ Wx^T + b.  M=8192, N=2048, K=2048.
// WG=256 thr (8 waves), tile 128M x 64N, K chunks of 64 double-buffered in LDS.
// LDS rows padded 64->68 dwords: WMMA fragment ds_load_b64 is conflict-free.
// ============================================================================
#define AM 128
#define AN 64
#define AKC 64
#define ASTR 68
#define A_SZ (AM * ASTR)        // 8704 dwords
#define B_SZ (AN * ASTR)        // 4352 dwords
#define CH_SZ (A_SZ + B_SZ)     // 13056 dwords per buffer
#define NCHUNK (H_DIM / AKC)    // 32
#define SMEM_A (2 * CH_SZ * 4)  // 104448 bytes

#if !USE_TDM
__device__ __forceinline__ void stageA_chunk(float* dA, float* dB, const float* x,
                                             const float* W, int mbase, int nbase,
                                             int k0, int tid) {
  for (int idx = tid; idx < AM * AKC; idx += 256) {
    int row = idx >> 6, col = idx & 63;
    dA[row * ASTR + col] = x[(size_t)(mbase + row) * H_DIM + k0 + col];
  }
  for (int idx = tid; idx < AN * AKC; idx += 256) {
    int row = idx >> 6, col = idx & 63;
    dB[row * ASTR + col] = W[(size_t)(nbase + row) * WROW2 + k0 + col];
  }
}
#endif

__launch_bounds__(256, 1)
__global__ void rnn_u_gemm_k(const float* __restrict__ x, const float* __restrict__ W,
                             const float* __restrict__ bias, float* __restrict__ out) {
  extern __shared__ float smem[];
  const int tid = threadIdx.x;
  const int lane = tid & 31;
  const int wv = tid >> 5;              // 0..7
  const int mbase = blockIdx.x * AM;    // 64 tiles
  const int nbase = blockIdx.y * AN;    // 32 tiles
  const int r = lane & 15;
  const int hh = lane >> 4;
  const int h2 = hh * 2;

  float bv[4];
#pragma unroll
  for (int a = 0; a < 4; ++a) bv[a] = bias[nbase + a * 16 + r];

  v8f acc[4] = {};

#if USE_TDM
  if (wv == 0) {  // wave-uniform; TDM ignores EXEC
    tdm_load_2d(0u, x + (size_t)mbase * H_DIM, AKC, AM, H_DIM, 5, 3);
    tdm_load_2d((uint32_t)(A_SZ * 4), W + (size_t)nbase * WROW2, AKC, AN, WROW2, 5, 3);
  }
#endif
  for (int i = 0; i < NCHUNK; ++i) {
    float* Abuf = smem + (i & 1) * CH_SZ;
    float* Bbuf = Abuf + A_SZ;
    __syncthreads();  // consumers of the other buffer are done -> safe to refill
#if USE_TDM
    if (wv == 0) {
      if (i + 1 < NCHUNK) {
        uint32_t boff = (uint32_t)(((i + 1) & 1) * CH_SZ * 4);
        tdm_load_2d(boff, x + (size_t)mbase * H_DIM + (size_t)(i + 1) * AKC,
                    AKC, AM, H_DIM, 5, 3);
        tdm_load_2d(boff + (uint32_t)(A_SZ * 4),
                    W + (size_t)nbase * WROW2 + (size_t)(i + 1) * AKC,
                    AKC, AN, WROW2, 5, 3);
        __builtin_amdgcn_s_wait_tensorcnt(2);  // chunk i complete (in-order TDM)
      } else {
        __builtin_amdgcn_s_wait_tensorcnt(0);
      }
    }
    __syncthreads();  // chunk i visible to all waves
#else
    stageA_chunk(Abuf, Bbuf, x, W, mbase, nbase, i * AKC, tid);
    __syncthreads();
#endif
    const float* ap = Abuf + (wv * 16 + r) * ASTR + h2;
    const float* bp = Bbuf + r * ASTR + h2;
#pragma unroll
    for (int k = 0; k < AKC; k += 4) {
      v2f av = *(const v2f*)(ap + k);
#pragma unroll
      for (int a = 0; a < 4; ++a) {
        v2f bw = *(const v2f*)(bp + a * 16 * ASTR + k);
        acc[a] = wmma_f32(av, bw, acc[a]);
      }
    }
  }
#pragma unroll
  for (int a = 0; a < 4; ++a) {
#pragma unroll
    for (int j = 0; j < 8; ++j) {
      size_t row = (size_t)(mbase + wv * 16 + j + 8 * hh);
      out[row * H_DIM + (size_t)(nbase + a * 16 + r)] = acc[a][j] + bv[a];
    }
  }
}

// ============================================================================
// Kernel B: scan.  out[:,t,:] += out[:,t-1,:] @ Wh^T, t = 1..511.
// 128 WGs x 4 waves. WG owns 16 output columns; K=2048 split 4-way over waves.
// Wh quarter (16x512) per wave lives in LDS for the whole scan (TDM once).
// h_{t-1} quarter staged per step via TDM. Row stride 512 -> 516 dwords via
// TDM pad (2 dwords / 256) => conflict-free ds_load_b64 fragments.
// Cross-WG sync: flags[t] release-arrive counter / acquire spin.
// ============================================================================
#define NWG_B 128
#define KQ 512
#define PROW 516
#define WBLK 8272                        // 16*516 + slack, dwords
#define RED_OFF (8 * WBLK)
#define SMEM_B ((8 * WBLK + 768) * 4)    // 267776 B  (ISA: up to 320KB/WG)

#if !USE_TDM
__device__ __forceinline__ void stage16xKQ(float* dst, const float* src,
                                           size_t rstride, int lane) {
  for (int idx = lane; idx < 16 * KQ; idx += 32) {
    int row = idx >> 9, col = idx & (KQ - 1);
    dst[row * PROW + col + ((col >> 8) << 1)] = src[(size_t)row * rstride + col];
  }
}
#endif

__launch_bounds__(128, 1)
__global__ void rnn_scan_k(const float* __restrict__ W, float* __restrict__ out,
                           float* __restrict__ hlast, int* flags) {
  extern __shared__ float smem[];
  const int tid = threadIdx.x;
  const int lane = tid & 31;
  const int wv = tid >> 5;           // 0..3 : K quarter
  const int g0 = blockIdx.x * 16;    // output-column tile
  const int r = lane & 15;
  const int hh = lane >> 4;
  const int h2 = hh * 2;

  float* whb = smem + (size_t)wv * WBLK;
  float* hbb = smem + (size_t)(4 + wv) * WBLK;
  float* red = smem + RED_OFF;

  // Stage this wave's Wh quarter once: Wh[g0+n][wv*512 + k], row stride 4096.
  const float* whsrc = W + (size_t)g0 * WROW2 + H_DIM + (size_t)wv * KQ;
#if USE_TDM
  tdm_load_2d((uint32_t)(wv * WBLK * 4), whsrc, KQ, 16, WROW2, 7, 1);
  __builtin_amdgcn_s_wait_tensorcnt(0);
#else
  stage16xKQ(whb, whsrc, WROW2, lane);
#endif

  for (int t = 1; t < T_DIM; ++t) {
    if (t > 1) {  // t==1: h_0 = out[:,0,:] written by kernel A (stream order)
      while (__hip_atomic_load(&flags[t - 1], __ATOMIC_ACQUIRE,
                               __HIP_MEMORY_SCOPE_AGENT) != NWG_B)
        __builtin_amdgcn_s_sleep(1);
    }
    // Stage h_{t-1}[m][wv*512 + k]; batch-row stride = T*H elements.
    const float* hsrc = out + (size_t)(t - 1) * H_DIM + (size_t)wv * KQ;
#if USE_TDM
    tdm_load_2d((uint32_t)((4 + wv) * WBLK * 4), hsrc, KQ, 16,
                (uint64_t)T_DIM * H_DIM, 7, 1);
    __builtin_amdgcn_s_wait_tensorcnt(0);
#else
    stage16xKQ(hbb, hsrc, (size_t)T_DIM * H_DIM, lane);
#endif
    v8f acc = {0.f, 0.f, 0.f, 0.f, 0.f, 0.f, 0.f, 0.f};
    const float* ap = hbb + r * PROW + h2;   // A: h_prev fragment (M=batch)
    const float* bp = whb + r * PROW + h2;   // B: Wh fragment   (N=out col)
#pragma unroll
    for (int k = 0; k < KQ; k += 4) {
      int ko = k + ((k >> 8) << 1);          // skip mid-row TDM pad
      v2f av = *(const v2f*)(ap + ko);
      v2f bw = *(const v2f*)(bp + ko);
      acc = wmma_f32(av, bw, acc);
    }
    // Cross-wave K reduction through LDS; wave 0 finalizes.
    if (wv != 0) {
#pragma unroll
      for (int j = 0; j < 8; ++j) red[(wv - 1) * 256 + lane * 8 + j] = acc[j];
    }
    __syncthreads();
    if (wv == 0) {
#pragma unroll
      for (int j = 0; j < 8; ++j) {
        float s = acc[j] + red[0 * 256 + lane * 8 + j] +
                  red[1 * 256 + lane * 8 + j] + red[2 * 256 + lane * 8 + j];
        size_t off = ((size_t)(j + 8 * hh) * T_DIM + (size_t)t) * H_DIM + g0 + r;
        float h = out[off] + s;              // out[off] currently holds u_t
        out[off] = h;
        if (t == T_DIM - 1)
          hlast[(size_t)(j + 8 * hh) * H_DIM + g0 + r] = h;
      }
      if (lane == 0)
        __hip_atomic_fetch_add(&flags[t], 1, __ATOMIC_RELEASE,
                               __HIP_MEMORY_SCOPE_AGENT);
    }
    __syncthreads();
  }
}

// ============================================================================
extern "C" void kernel_launch(void* const* d_in, const int* in_sizes, int n_in,
                              void* d_out, int out_size, void* d_ws, size_t ws_size,
                              hipStream_t stream) {
  (void)in_sizes; (void)n_in; (void)out_size; (void)ws_size;
  const float* x = (const float*)d_in[0];
  const float* W = (const float*)d_in[1];
  const float* b = (const float*)d_in[2];
  float* out = (float*)d_out;
  float* hlast = out + (size_t)B_DIM * T_DIM * H_DIM;
  int* flags = (int*)d_ws;

  // Allow >64KB dynamic LDS (host-side attribute set; deterministic each call).
  (void)hipFuncSetAttribute((const void*)rnn_u_gemm_k,
                            hipFuncAttributeMaxDynamicSharedMemorySize, SMEM_A);
  (void)hipFuncSetAttribute((const void*)rnn_scan_k,
                            hipFuncAttributeMaxDynamicSharedMemorySize, SMEM_B);

  (void)hipMemsetAsync(flags, 0, T_DIM * sizeof(int), stream);

  dim3 gA(B_DIM * T_DIM / AM, H_DIM / AN, 1);  // 64 x 32
  rnn_u_gemm_k<<<gA, 256, SMEM_A, stream>>>(x, W, b, out);
  rnn_scan_k<<<NWG_B, 128, SMEM_B, stream>>>(W, out, hlast, flags);
}